// Combi_Layer_26955214750057
// MI455X (gfx1250) — compile-verified
//
#include <hip/hip_runtime.h>

// ---------------------------------------------------------------------------
// Combi layer (FNO + diff conv) for MI455X / gfx1250, wave32.
//
// Spectral branch = truncated DFT GEMMs (only 64x32 modes nonzero), all on
// V_WMMA_F32_16X16X4_F32.  Stationary GEMM operands (DFT matrices, out_ft
// tiles) are staged LDS-side by the Tensor Data Mover (tensor_load_to_lds,
// TENSORcnt) with hardware row padding; moving operands use the vector path.
// Memory-bound: ~1.1 GB traffic vs ~20 GFLOP -> fp32 WMMA is free and exact.
// ---------------------------------------------------------------------------

typedef __attribute__((ext_vector_type(2))) float        v2f;
typedef __attribute__((ext_vector_type(8))) float        v8f;
typedef __attribute__((ext_vector_type(4))) unsigned int v4u;
typedef __attribute__((ext_vector_type(4))) int          v4i;
typedef __attribute__((ext_vector_type(8))) int          v8i;

__device__ __forceinline__ v8f wmma4(v2f a, v2f b, v8f c) {
  // D(16x16,f32) = A(16x4,f32) * B(4x16,f32) + C
  return __builtin_amdgcn_wmma_f32_16x16x4_f32(
      false, a, false, b, (short)0, c, false, false);
}

// ---- Tensor Data Mover: 2D tile (global -> LDS) with hardware padding ------
// tile0 elements per row (4B each), tile1 rows, stride0 = row stride in
// elements.  padIntCode: pad every 2^(padIntCode+1) dwords; padAmtCode:
// insert (padAmtCode+1) dwords of padding.  Issued once per calling wave.
__device__ __forceinline__ void tdm_load_2d(unsigned lds_addr, const void* gptr,
                                            unsigned tile0, unsigned tile1,
                                            unsigned stride0,
                                            unsigned padIntCode,
                                            unsigned padAmtCode) {
  unsigned long long ga = (unsigned long long)(size_t)gptr;
  const unsigned dim0 = 1u << 30, dim1 = 1u << 30;   // no OOB clipping
  v4u g0;
  g0.x = 1u;                                         // count=1 (valid D#)
  g0.y = lds_addr;                                   // LDS byte address
  g0.z = (unsigned)(ga & 0xFFFFFFFFu);               // global_addr[31:0]
  g0.w = (unsigned)((ga >> 32) & 0x01FFFFFFu) | (2u << 30); // [56:32] | type=2
  v8i g1;
  g1[0] = (int)((2u << 16)                           // data_size = 4 bytes
              | (1u << 20)                           // pad_enable
              | (padIntCode << 22) | (padAmtCode << 25));
  g1[1] = (int)((dim0 & 0xFFFFu) << 16);             // tensor_dim0 lo16
  g1[2] = (int)((dim0 >> 16) | ((dim1 & 0xFFFFu) << 16));
  g1[3] = (int)((dim1 >> 16) | (tile0 << 16));       // tile_dim0
  g1[4] = (int)(tile1 & 0xFFFFu);                    // tile_dim1 (tile_dim2=0)
  g1[5] = (int)stride0;                              // dim0_stride lo32
  g1[6] = 0;
  g1[7] = 0;
  v4i z4 = {0, 0, 0, 0};
#if __clang_major__ >= 23
  v8i z8 = {};
  __builtin_amdgcn_tensor_load_to_lds(g0, g1, z4, z4, z8, 0);
#else
  __builtin_amdgcn_tensor_load_to_lds(g0, g1, z4, z4, 0);
#endif
}

// low 32 bits of a generic pointer to a __shared__ object = LDS byte offset
__device__ __forceinline__ unsigned lds_of(const void* p) {
  return (unsigned)(size_t)p;
}

#define B_  16
#define CI  32
#define CO  32
#define H_  256
#define W_  256

// workspace layout (float offsets)
#define WS_FA 0               // [256 w][64 n]  forward W-DFT (cos | -sin)
#define WS_ER 16384           // [64 kx][256 h] forward H-DFT real
#define WS_EI 32768           // [64 kx][256 h] forward H-DFT imag
#define WS_GR 49152           // [256 h][64 kx] inverse H-DFT real (/H)
#define WS_GI 65536           // [256 h][64 kx] inverse H-DFT imag (/H)
#define WS_FE 81920           // [64 k][256 w]  inverse W transform (irfft fold)
#define WS_Y  98304           // [131072 rows][64]  Y (stages A,B) then T (D,E)
#define WS_ZR 8486912         // [512 planes][2048] Z real -> out_ft real
#define WS_ZI 9535488         // [512 planes][2048] Z imag -> out_ft imag
// total = 10,584,064 floats (~40.4 MB of d_ws)

// ---------------------------------------------------------------------------
// Twiddle generation (phases reduced mod 256 -> exact periodic angles)
// ---------------------------------------------------------------------------
__global__ void k_twiddles(float* __restrict__ ws) {
  const float TH = 6.28318530717958647692f / 256.0f;
  const int total = 6 * 16384;
  for (int i = blockIdx.x * blockDim.x + threadIdx.x; i < total;
       i += gridDim.x * blockDim.x) {
    int seg = i >> 14, j = i & 16383;
    if (seg == 0) {                       // FA[w][n]
      int w = j >> 6, n = j & 63, ky = n & 31;
      float ang = TH * (float)((w * ky) & 255);
      ws[WS_FA + j] = (n < 32) ? __cosf(ang) : -__sinf(ang);
    } else if (seg <= 2) {                // Er / Ei  [kx][h]
      int kx = j >> 8, hh = j & 255;
      int kxp = (kx < 32) ? kx : kx + 192;
      float ang = TH * (float)((kxp * hh) & 255);
      if (seg == 1) ws[WS_ER + j] = __cosf(ang);
      else          ws[WS_EI + j] = -__sinf(ang);
    } else if (seg <= 4) {                // Gr / Gi  [h][kx], scaled 1/H
      int hh = j >> 6, kx = j & 63;
      int kxp = (kx < 32) ? kx : kx + 192;
      float ang = TH * (float)((kxp * hh) & 255);
      if (seg == 3) ws[WS_GR + j] = __cosf(ang) * (1.0f / 256.0f);
      else          ws[WS_GI + j] = __sinf(ang) * (1.0f / 256.0f);
    } else {                              // FE[k][w] irfft-W fold, scaled 1/W
      int k = j >> 8, w = j & 255, ky = k & 31;
      float ang = TH * (float)((ky * w) & 255);
      float coef = (ky == 0) ? (1.0f / 256.0f) : (2.0f / 256.0f);
      float v;
      if (k < 32) v = coef * __cosf(ang);
      else        v = (ky == 0) ? 0.0f : -coef * __sinf(ang);
      ws[WS_FE + j] = v;
    }
  }
}

// ---------------------------------------------------------------------------
// Diff branch: out[b,o,h,w] = sum_k conv_w[o,k]*feats[k] + bias[o].
// feats = [x | dh | dw] built branch-free in LDS at staging time; the WMMA
// K-loop is then pure ds_load -> v_wmma (no per-lane guards).
// Block = (b, h, half-row of 128 pixels).  Tile: M=o(16) x N=pix(16), K=96.
// ---------------------------------------------------------------------------
__global__ __launch_bounds__(256) void k_diff(const float* __restrict__ x,
                                              const float* __restrict__ cw,
                                              const float* __restrict__ cb,
                                              float* __restrict__ out) {
  const int bi = blockIdx.x;            // 8192 = B*H*2
  const int w0 = (bi & 1) * 128;
  const int h  = (bi >> 1) & 255;
  const int b  = bi >> 9;

  __shared__ float sf[96][130];         // [x | dh | dw], 129 cols used
  __shared__ float scw[CO][100];        // conv_w [o][96]
  __shared__ float sb[CO];

  const int tid = threadIdx.x;
  // pass 1: x row h, and dh = x[h+1]-x[h] (0 on last row)
  for (int idx = tid; idx < CI * 130; idx += 256) {
    int c = idx / 130, j = idx % 130;
    int w = w0 + j; if (w > 255) w = 255;
    size_t base = ((size_t)(b * CI + c) * H_ + h) * W_ + w;
    float xc = x[base];
    sf[c][j] = xc;
    sf[32 + c][j] = (h < 255) ? (x[base + W_] - xc) : 0.0f;
  }
  for (int idx = tid; idx < CO * 96; idx += 256)
    scw[idx / 96][idx % 96] = cw[idx];
  if (tid < CO) sb[tid] = cb[tid];
  __syncthreads();
  // pass 2: dw = x[w+1]-x[w] from LDS (0 at w=255)
  for (int idx = tid; idx < CI * 128; idx += 256) {
    int c = idx >> 7, j = idx & 127;
    sf[64 + c][j] = ((w0 + j) < 255) ? (sf[c][j + 1] - sf[c][j]) : 0.0f;
  }
  __syncthreads();

  const int lane = tid & 31, wv = tid >> 5;
  const int lr = lane & 15, lh = lane >> 4;

  for (int it = 0; it < 2; ++it) {
    int t = wv + 8 * it;                // 16 tiles: 2 o-tiles x 8 pixel-tiles
    int ot = t >> 3, wt = t & 7;
    int p  = wt * 16 + lr;
    v8f acc = {};
    for (int kk = 0; kk < 24; ++kk) {   // K = 96
      int k0 = kk * 4 + lh * 2;
      v2f a, bb;
      a.x  = scw[ot * 16 + lr][k0];
      a.y  = scw[ot * 16 + lr][k0 + 1];
      bb.x = sf[k0][p];
      bb.y = sf[k0 + 1][p];
      acc = wmma4(a, bb, acc);
    }
    for (int v = 0; v < 8; ++v) {
      int o = ot * 16 + v + 8 * lh;
      out[((size_t)(b * CO + o) * H_ + h) * W_ + w0 + p] = acc[v] + sb[o];
    }
  }
}

// ---------------------------------------------------------------------------
// Stage A: W-direction truncated DFT.  GEMM M=131072 rows (b,c,h), K=256,
// N=64 (Re|Im).  FA chunks staged by TDM; x rows by the vector path.
// ---------------------------------------------------------------------------
__global__ __launch_bounds__(256) void k_stageA(const float* __restrict__ x,
                                                float* __restrict__ ws) {
  const int rbase = blockIdx.x * 32;    // 4096 blocks
  const float* FA = ws + WS_FA;
  float* Y = ws + WS_Y;
  __shared__ float sA[32][66];
  __shared__ float sB[64][66];
  const int tid = threadIdx.x, lane = tid & 31, wv = tid >> 5;
  const int lr = lane & 15, lh = lane >> 4;
  const int mt = wv >> 2, nt = wv & 3;  // 2 x 4 tiles, one per wave
  v8f acc = {};
  for (int ch = 0; ch < 4; ++ch) {      // K chunks of 64
    __syncthreads();
    if (wv == 0)                        // FA rows ch*64.. : 64x64, stride 64
      tdm_load_2d(lds_of(&sB[0][0]), FA + (size_t)ch * 64 * 64,
                  64, 64, 64, /*pad every 64 dw*/ 5, /*+2 dw*/ 1);
    for (int idx = tid; idx < 32 * 64; idx += 256) {
      int r = idx >> 6, j = idx & 63;
      sA[r][j] = x[(size_t)(rbase + r) * 256 + ch * 64 + j];
    }
    if (wv == 0) __builtin_amdgcn_s_wait_tensorcnt(0);
    __syncthreads();
    for (int kk = 0; kk < 16; ++kk) {
      int k0 = kk * 4 + lh * 2;
      v2f a, b;
      a.x = sA[mt * 16 + lr][k0];  a.y = sA[mt * 16 + lr][k0 + 1];
      b.x = sB[k0][nt * 16 + lr];  b.y = sB[k0 + 1][nt * 16 + lr];
      acc = wmma4(a, b, acc);
    }
  }
  for (int v = 0; v < 8; ++v) {
    int m = mt * 16 + v + 8 * lh;
    Y[(size_t)(rbase + m) * 64 + nt * 16 + lr] = acc[v];
  }
}

// ---------------------------------------------------------------------------
// Stage B: Z[kx,ky] = sum_h E[kx,h] * Y[h,ky]  (complex), per (b,c).
// E chunks staged by TDM, Y chunks by vector path.
// ---------------------------------------------------------------------------
__global__ __launch_bounds__(256) void k_stageB(float* __restrict__ ws) {
  const int bc = blockIdx.x;            // 512
  const float* Er = ws + WS_ER;
  const float* Ei = ws + WS_EI;
  const float* Y  = ws + WS_Y;
  __shared__ float sY[64][66];
  __shared__ float sEr[64][66];
  __shared__ float sEi[64][66];
  const int tid = threadIdx.x, lane = tid & 31, wv = tid >> 5;
  const int lr = lane & 15, lh = lane >> 4;
  const int mt = wv >> 1, nt = wv & 1;  // 4 kx-tiles x 2 ky-tiles
  v8f P = {}, Q = {}, R = {}, S = {};
  for (int ch = 0; ch < 4; ++ch) {      // h chunks of 64
    __syncthreads();
    if (wv == 0) {                      // E cols ch*64.. : 64x64, stride 256
      tdm_load_2d(lds_of(&sEr[0][0]), Er + ch * 64, 64, 64, 256, 5, 1);
      tdm_load_2d(lds_of(&sEi[0][0]), Ei + ch * 64, 64, 64, 256, 5, 1);
    }
    for (int idx = tid; idx < 64 * 64; idx += 256) {
      int hh = idx >> 6, n = idx & 63;
      sY[hh][n] = Y[((size_t)bc * 256 + ch * 64 + hh) * 64 + n];
    }
    if (wv == 0) __builtin_amdgcn_s_wait_tensorcnt(0);
    __syncthreads();
    for (int kk = 0; kk < 16; ++kk) {
      int k0 = kk * 4 + lh * 2;
      v2f ar, ai, br, bi;
      ar.x = sEr[mt * 16 + lr][k0];  ar.y = sEr[mt * 16 + lr][k0 + 1];
      ai.x = sEi[mt * 16 + lr][k0];  ai.y = sEi[mt * 16 + lr][k0 + 1];
      br.x = sY[k0][nt * 16 + lr];       br.y = sY[k0 + 1][nt * 16 + lr];
      bi.x = sY[k0][32 + nt * 16 + lr];  bi.y = sY[k0 + 1][32 + nt * 16 + lr];
      P = wmma4(ar, br, P);
      Q = wmma4(ai, bi, Q);
      R = wmma4(ar, bi, R);
      S = wmma4(ai, br, S);
    }
  }
  float* Zr = ws + WS_ZR;
  float* Zi = ws + WS_ZI;
  for (int v = 0; v < 8; ++v) {
    int kx = mt * 16 + v + 8 * lh;
    int ky = nt * 16 + lr;
    size_t f = (size_t)bc * 2048 + kx * 32 + ky;
    Zr[f] = P[v] - Q[v];
    Zi[f] = R[v] + S[v];
  }
}

// ---------------------------------------------------------------------------
// Stage C: per-frequency complex mix  O[b,o] = sum_i Z[b,i] * W[i,o].
// Block = 4 freqs; out_ft overwrites Z planes in place (same freqs; all data
// staged into LDS before any write -> race-free).
// ---------------------------------------------------------------------------
__global__ __launch_bounds__(256) void k_stageC(const float* __restrict__ w1r,
                                                const float* __restrict__ w1i,
                                                const float* __restrict__ w2r,
                                                const float* __restrict__ w2i,
                                                float* __restrict__ ws) {
  const int fbase = blockIdx.x * 4;     // 512 blocks -> 2048 freqs
  float* Zr = ws + WS_ZR;
  float* Zi = ws + WS_ZI;
  __shared__ float sZr[4][16][34];
  __shared__ float sZi[4][16][34];
  __shared__ float sWr[4][32][34];
  __shared__ float sWi[4][32][34];
  const int tid = threadIdx.x;
  for (int idx = tid; idx < 4 * 512; idx += 256) {
    int q = idx >> 9, r = idx & 511;
    int b = r >> 5, i = r & 31;
    size_t src = (size_t)(b * CI + i) * 2048 + fbase + q;
    sZr[q][b][i] = Zr[src];
    sZi[q][b][i] = Zi[src];
  }
  for (int idx = tid; idx < 4 * 1024; idx += 256) {
    int q = idx >> 10, r = idx & 1023;
    int i = r >> 5, o = r & 31;
    int f = fbase + q, kx = f >> 5, ky = f & 31;
    size_t src;
    if (kx < 32) {
      src = (size_t)(i * CO + o) * 1024 + kx * 32 + ky;
      sWr[q][i][o] = w1r[src];  sWi[q][i][o] = w1i[src];
    } else {
      src = (size_t)(i * CO + o) * 1024 + (kx - 32) * 32 + ky;
      sWr[q][i][o] = w2r[src];  sWi[q][i][o] = w2i[src];
    }
  }
  __syncthreads();
  const int lane = tid & 31, wv = tid >> 5;
  const int lr = lane & 15, lh = lane >> 4;
  const int q = wv >> 1, nt = wv & 1;   // (freq, o-tile) per wave
  v8f P = {}, Q = {}, R = {}, S = {};
  for (int kk = 0; kk < 8; ++kk) {      // K = 32 (input channels)
    int k0 = kk * 4 + lh * 2;
    v2f azr, azi, bwr, bwi;
    azr.x = sZr[q][lr][k0];  azr.y = sZr[q][lr][k0 + 1];
    azi.x = sZi[q][lr][k0];  azi.y = sZi[q][lr][k0 + 1];
    bwr.x = sWr[q][k0][nt * 16 + lr];  bwr.y = sWr[q][k0 + 1][nt * 16 + lr];
    bwi.x = sWi[q][k0][nt * 16 + lr];  bwi.y = sWi[q][k0 + 1][nt * 16 + lr];
    P = wmma4(azr, bwr, P);
    Q = wmma4(azi, bwi, Q);
    R = wmma4(azr, bwi, R);
    S = wmma4(azi, bwr, S);
  }
  int f = fbase + q;
  for (int v = 0; v < 8; ++v) {
    int b = v + 8 * lh;                 // batch (M=16)
    int o = nt * 16 + lr;
    size_t dst = (size_t)(b * CO + o) * 2048 + f;
    Zr[dst] = P[v] - Q[v];              // out_ft real (in place)
    Zi[dst] = R[v] + S[v];              // out_ft imag
  }
}

// ---------------------------------------------------------------------------
// Stage D: tmp[h,ky] = (1/H) sum_kx out_ft[kx,ky] e^{+i...}, per (b,o).
// out_ft tile + G chunks staged by TDM.  Writes T rows (reuses Y region).
// ---------------------------------------------------------------------------
__global__ __launch_bounds__(256) void k_stageD(float* __restrict__ ws) {
  const int bo = blockIdx.x;            // 512
  const float* Gr = ws + WS_GR;
  const float* Gi = ws + WS_GI;
  const float* Or = ws + WS_ZR;
  const float* Oi = ws + WS_ZI;
  float* T = ws + WS_Y;
  __shared__ float sOr[64][34];
  __shared__ float sOi[64][34];
  __shared__ float sGr[64][66];
  __shared__ float sGi[64][66];
  const int tid = threadIdx.x, lane = tid & 31, wv = tid >> 5;
  const int lr = lane & 15, lh = lane >> 4;
  if (wv == 0) {                        // out_ft tile: 64x32, contiguous
    tdm_load_2d(lds_of(&sOr[0][0]), Or + (size_t)bo * 2048,
                32, 64, 32, /*pad every 32 dw*/ 4, /*+2 dw*/ 1);
    tdm_load_2d(lds_of(&sOi[0][0]), Oi + (size_t)bo * 2048,
                32, 64, 32, 4, 1);
  }
  for (int hc = 0; hc < 4; ++hc) {      // output h chunks of 64
    __syncthreads();
    if (wv == 0) {                      // G rows hc*64.. : 64x64, stride 64
      tdm_load_2d(lds_of(&sGr[0][0]), Gr + (size_t)hc * 64 * 64,
                  64, 64, 64, 5, 1);
      tdm_load_2d(lds_of(&sGi[0][0]), Gi + (size_t)hc * 64 * 64,
                  64, 64, 64, 5, 1);
      __builtin_amdgcn_s_wait_tensorcnt(0);
    }
    __syncthreads();
    const int mt = wv >> 1, nt = wv & 1;  // 4 h-tiles x 2 ky-tiles per chunk
    v8f P = {}, Q = {}, R = {}, S = {};
    for (int kk = 0; kk < 16; ++kk) {     // K = 64 (kx)
      int k0 = kk * 4 + lh * 2;
      v2f ar, ai, br, bi;
      ar.x = sGr[mt * 16 + lr][k0];  ar.y = sGr[mt * 16 + lr][k0 + 1];
      ai.x = sGi[mt * 16 + lr][k0];  ai.y = sGi[mt * 16 + lr][k0 + 1];
      br.x = sOr[k0][nt * 16 + lr];  br.y = sOr[k0 + 1][nt * 16 + lr];
      bi.x = sOi[k0][nt * 16 + lr];  bi.y = sOi[k0 + 1][nt * 16 + lr];
      P = wmma4(ar, br, P);
      Q = wmma4(ai, bi, Q);
      R = wmma4(ar, bi, R);
      S = wmma4(ai, br, S);
    }
    for (int v = 0; v < 8; ++v) {
      int h  = hc * 64 + mt * 16 + v + 8 * lh;
      int ky = nt * 16 + lr;
      size_t row = (size_t)bo * 256 + h;
      T[row * 64 + ky]      = P[v] - Q[v];
      T[row * 64 + 32 + ky] = R[v] + S[v];
    }
  }
}

// ---------------------------------------------------------------------------
// Stage E: out[r,w] += sum_k T[r,k] * FE[k,w]  (irfft along W, folded).
// GEMM M=131072, K=64, N=256.  FE chunk staged by TDM.
// ---------------------------------------------------------------------------
__global__ __launch_bounds__(256) void k_stageE(float* __restrict__ ws,
                                                float* __restrict__ out) {
  const int bi = blockIdx.x;            // 4096
  const int rbase = (bi >> 1) * 64;
  const int wc = (bi & 1) * 128;
  const float* T  = ws + WS_Y;
  const float* FE = ws + WS_FE;
  __shared__ float sT[64][66];
  __shared__ float sF[64][130];
  const int tid = threadIdx.x, lane = tid & 31, wv = tid >> 5;
  const int lr = lane & 15, lh = lane >> 4;
  if (wv == 0)                          // FE cols wc.. : 128x64, stride 256
    tdm_load_2d(lds_of(&sF[0][0]), FE + wc,
                128, 64, 256, /*pad every 128 dw*/ 6, /*+2 dw*/ 1);
  for (int idx = tid; idx < 64 * 64; idx += 256) {
    int r = idx >> 6, k = idx & 63;
    sT[r][k] = T[(size_t)(rbase + r) * 64 + k];
  }
  if (wv == 0) __builtin_amdgcn_s_wait_tensorcnt(0);
  __syncthreads();
  for (int it = 0; it < 4; ++it) {
    int t = wv + 8 * it;                // 32 tiles: 4 row-tiles x 8 col-tiles
    int mt = t >> 3, nt = t & 7;
    v8f acc = {};
    for (int kk = 0; kk < 16; ++kk) {   // K = 64
      int k0 = kk * 4 + lh * 2;
      v2f a, b;
      a.x = sT[mt * 16 + lr][k0];  a.y = sT[mt * 16 + lr][k0 + 1];
      b.x = sF[k0][nt * 16 + lr];  b.y = sF[k0 + 1][nt * 16 + lr];
      acc = wmma4(a, b, acc);
    }
    for (int v = 0; v < 8; ++v) {
      int m = mt * 16 + v + 8 * lh;
      size_t addr = (size_t)(rbase + m) * 256 + wc + nt * 16 + lr;
      out[addr] += acc[v];              // diff branch already wrote out
    }
  }
}

// ---------------------------------------------------------------------------
extern "C" void kernel_launch(void* const* d_in, const int* in_sizes, int n_in,
                              void* d_out, int out_size, void* d_ws, size_t ws_size,
                              hipStream_t stream) {
  const float* x   = (const float*)d_in[0];
  const float* cw  = (const float*)d_in[1];
  const float* cb  = (const float*)d_in[2];
  const float* w1r = (const float*)d_in[3];
  const float* w1i = (const float*)d_in[4];
  const float* w2r = (const float*)d_in[5];
  const float* w2i = (const float*)d_in[6];
  float* out = (float*)d_out;
  float* ws  = (float*)d_ws;   // needs ~40.4 MB

  k_twiddles<<<96, 256, 0, stream>>>(ws);
  k_diff   <<<8192, 256, 0, stream>>>(x, cw, cb, out);   // writes d_out
  k_stageA <<<4096, 256, 0, stream>>>(x, ws);
  k_stageB <<<512,  256, 0, stream>>>(ws);
  k_stageC <<<512,  256, 0, stream>>>(w1r, w1i, w2r, w2i, ws);
  k_stageD <<<512,  256, 0, stream>>>(ws);
  k_stageE <<<4096, 256, 0, stream>>>(ws, out);          // += spectral branch
}